// Convolutionlayer_20684562498156
// MI455X (gfx1250) — compile-verified
//
#include <hip/hip_runtime.h>

typedef float v2f __attribute__((ext_vector_type(2)));
typedef float v8f __attribute__((ext_vector_type(8)));
typedef unsigned int v4u __attribute__((ext_vector_type(4)));
typedef int v8i __attribute__((ext_vector_type(8)));
typedef int v4i __attribute__((ext_vector_type(4)));

#define B_    32
#define N_    1024
#define FIN_  128
#define FOUT_ 128
#define KB    32            // K-block staged per stage
#define ROWS  64            // output rows per block
#define ADJ_STRIDE 36       // KB + 4 pad dwords (TDM pad reproduces this)
#define AGG_STRIDE 132      // 128 + 4 pad

// LDS layout in floats (double-buffered adj + node tiles; agg overlays later)
#define ADJ0_OFF   0
#define ADJ1_OFF   (ROWS * ADJ_STRIDE)                  // 2304
#define NODE0_OFF  (2 * ROWS * ADJ_STRIDE)              // 4608
#define NODE1_OFF  (2 * ROWS * ADJ_STRIDE + KB * FIN_)  // 8704
#define LDS_FLOATS (2 * ROWS * ADJ_STRIDE + 2 * KB * FIN_) // 12800 (50 KB)

// ---------------------------------------------------------------------------
// Kernel 1: per-row adjacency sums -> 1/deg  (one wave32 per row)
// ---------------------------------------------------------------------------
__global__ __launch_bounds__(256)
void rowsum_inv_kernel(const float* __restrict__ adj, float* __restrict__ invdeg) {
    const int wave = threadIdx.x >> 5;
    const int lane = threadIdx.x & 31;
    const long long row = (long long)blockIdx.x * 8 + wave;   // 0 .. B*N-1
    const float* p = adj + row * N_;
    float s = 0.f;
    for (int m = lane; m < N_; m += 32) s += p[m];
#pragma unroll
    for (int off = 16; off > 0; off >>= 1)
        s += __shfl_xor(s, off, 32);
    if (lane == 0) invdeg[row] = 1.0f / s;
}

// ---------------------------------------------------------------------------
// TDM: issue a 2D tensor_load_to_lds per CDNA5 ISA §8 (D# groups in SGPRs).
//  tile_d0 = contiguous elements per row, tile_d1 = rows,
//  stride0 = row stride of the global tensor (elements, fp32),
//  optional LDS padding: pad_int code (4 -> every 32 DWORDs), pad_amt code
//  (3 -> 4 DWORDs) turns a 32-float row into a 36-float LDS stride.
// ---------------------------------------------------------------------------
__device__ __forceinline__ void tdm_load_2d(unsigned lds_off_bytes,
                                            const float* gptr,
                                            unsigned tile_d0, unsigned tile_d1,
                                            unsigned stride0_elts,
                                            unsigned pad_en, unsigned pad_int,
                                            unsigned pad_amt) {
    const unsigned long long ga = (unsigned long long)(const void*)gptr;
    v4u g0;
    g0[0] = 1u;                                        // count = 1, user mode
    g0[1] = lds_off_bytes;                             // lds_addr
    g0[2] = (unsigned)ga;                              // global_addr[31:0]
    g0[3] = (unsigned)((ga >> 32) & 0x01FFFFFFu) | 0x80000000u; // addr[56:32] | type=2

    const unsigned td0 = stride0_elts;                 // tensor_dim0 (in-bounds)
    v8i g1;
    g1[0] = (int)((2u << 16) | (pad_en << 20) | (pad_int << 22) | (pad_amt << 25));
    g1[1] = (int)((td0 & 0xFFFFu) << 16);              // tensor_dim0[15:0]
    g1[2] = (int)((td0 >> 16) & 0xFFFFu);              // dim0[31:16] | dim1 lo16 (=0)
    g1[3] = (int)(16u | (tile_d0 << 16));              // tensor_dim1 = 1<<20 | tile_dim0
    g1[4] = (int)(tile_d1 & 0xFFFFu);                  // tile_dim1, tile_dim2 = 0
    g1[5] = (int)stride0_elts;                         // tensor_dim0_stride[31:0]
    g1[6] = 0;                                         // stride0[47:32] | stride1 lo
    g1[7] = 0;
    const v4i z4 = {0, 0, 0, 0};
    const v8i z8 = {0, 0, 0, 0, 0, 0, 0, 0};
    __builtin_amdgcn_tensor_load_to_lds(g0, g1, z4, z4, z8, 0);
}

__device__ __forceinline__ v8f wmma4(v2f a, v2f b, v8f c) {
    return __builtin_amdgcn_wmma_f32_16x16x4_f32(false, a, false, b,
                                                 (short)0, c, false, false);
}

// ---------------------------------------------------------------------------
// Kernel 2: fused  out = LeakyReLU( diag(invdeg) * (A @ X) @ W^T + b )
// 256 threads / 8 waves; block tile = 64 rows x 128 cols.
// Wave w: rows base (w>>2)*32 (2 M-tiles), cols (w&3)*32 (2 N-tiles)
//   -> 4 WMMAs per K-step from 2 A-frags + 2 B-frags.
// Staging: TDM double-buffer issued by wave 0, synced with s_wait_tensorcnt.
// ---------------------------------------------------------------------------
__global__ __launch_bounds__(256)
void gcn_fused_kernel(const float* __restrict__ node,
                      const float* __restrict__ adj,
                      const float* __restrict__ W,
                      const float* __restrict__ bias,
                      const float* __restrict__ invdeg,
                      float* __restrict__ out) {
    __shared__ __align__(16) float s_mem[LDS_FLOATS];

    const int tid  = threadIdx.x;
    const int wave = tid >> 5;
    const int lane = tid & 31;
    const int half = lane >> 4;
    const int lrow = lane & 15;

    const int blocksPerBatch = N_ / ROWS;              // 16
    const int batch   = blockIdx.x / blocksPerBatch;
    const int rowbase = (blockIdx.x % blocksPerBatch) * ROWS;

    const float* adjB  = adj  + ((long long)batch * N_ + rowbase) * N_;
    const float* nodeB = node + (long long)batch * N_ * FIN_;
    const unsigned ldsBase = (unsigned)(unsigned long long)(void*)s_mem;

    const int mbase = (wave >> 2) * 32;
    const int ncol  = (wave & 3) * 32;

    v8f acc00 = {}, acc01 = {}, acc10 = {}, acc11 = {};
    const int S = N_ / KB;                             // 32 stages

    if (wave == 0) {   // prologue: stage 0 into buffer 0
        tdm_load_2d(ldsBase + ADJ0_OFF * 4, adjB, KB, ROWS, N_, 1u, 4u, 3u);
        tdm_load_2d(ldsBase + NODE0_OFF * 4, nodeB, FIN_, KB, FIN_, 0u, 0u, 0u);
    }

    for (int s = 0; s < S; ++s) {
        const int buf = s & 1;
        if (wave == 0) {
            if (s + 1 < S) {   // prefetch next stage into other buffer
                const int nb = (s + 1) & 1;
                tdm_load_2d(ldsBase + (nb ? ADJ1_OFF : ADJ0_OFF) * 4,
                            adjB + (s + 1) * KB, KB, ROWS, N_, 1u, 4u, 3u);
                tdm_load_2d(ldsBase + (nb ? NODE1_OFF : NODE0_OFF) * 4,
                            nodeB + (long long)(s + 1) * KB * FIN_,
                            FIN_, KB, FIN_, 0u, 0u, 0u);
                __builtin_amdgcn_s_wait_tensorcnt(2);  // current stage landed
            } else {
                __builtin_amdgcn_s_wait_tensorcnt(0);
            }
        }
        __syncthreads();

        const float* sadj  = s_mem + (buf ? ADJ1_OFF : ADJ0_OFF);
        const float* snode = s_mem + (buf ? NODE1_OFF : NODE0_OFF);
#pragma unroll
        for (int kk = 0; kk < KB; kk += 4) {
            const int kh = kk + 2 * half;
            v2f a0, a1, b0, b1;
            a0.x = sadj[(mbase + lrow) * ADJ_STRIDE + kh];
            a0.y = sadj[(mbase + lrow) * ADJ_STRIDE + kh + 1];
            a1.x = sadj[(mbase + 16 + lrow) * ADJ_STRIDE + kh];
            a1.y = sadj[(mbase + 16 + lrow) * ADJ_STRIDE + kh + 1];
            b0.x = snode[kh * FIN_ + ncol + lrow];
            b0.y = snode[(kh + 1) * FIN_ + ncol + lrow];
            b1.x = snode[kh * FIN_ + ncol + 16 + lrow];
            b1.y = snode[(kh + 1) * FIN_ + ncol + 16 + lrow];
            acc00 = wmma4(a0, b0, acc00);
            acc01 = wmma4(a0, b1, acc01);
            acc10 = wmma4(a1, b0, acc10);
            acc11 = wmma4(a1, b1, acc11);
        }
        __syncthreads();
    }

    // ---- scale rows by 1/deg; park agg tile (64 x 128, stride 132) in LDS ----
    const long long gIdx = (long long)batch * N_ + rowbase;
    float invd0[8], invd1[8];
#pragma unroll
    for (int v = 0; v < 8; ++v) {
        invd0[v] = invdeg[gIdx + mbase + 8 * half + v];
        invd1[v] = invdeg[gIdx + mbase + 16 + 8 * half + v];
    }
    float* sagg = s_mem;   // overlays staging buffers (all waves past barrier)
#pragma unroll
    for (int v = 0; v < 8; ++v) {
        const int r0 = mbase + v + 8 * half;
        sagg[r0 * AGG_STRIDE + ncol + lrow]             = acc00[v] * invd0[v];
        sagg[r0 * AGG_STRIDE + ncol + 16 + lrow]        = acc01[v] * invd0[v];
        sagg[(r0 + 16) * AGG_STRIDE + ncol + lrow]      = acc10[v] * invd1[v];
        sagg[(r0 + 16) * AGG_STRIDE + ncol + 16 + lrow] = acc11[v] * invd1[v];
    }
    __syncthreads();

    // ---- GEMM2: out_tile = agg @ W^T  (W: [FOUT, FIN], L2-resident 64 KB) ----
    v8f o00 = {}, o01 = {}, o10 = {}, o11 = {};
    const int oc = ncol + lrow;
#pragma unroll 8
    for (int kf = 0; kf < FIN_; kf += 4) {
        const int kh = kf + 2 * half;
        v2f a0, a1, b0, b1;
        a0.x = sagg[(mbase + lrow) * AGG_STRIDE + kh];
        a0.y = sagg[(mbase + lrow) * AGG_STRIDE + kh + 1];
        a1.x = sagg[(mbase + 16 + lrow) * AGG_STRIDE + kh];
        a1.y = sagg[(mbase + 16 + lrow) * AGG_STRIDE + kh + 1];
        b0.x = W[oc * FIN_ + kh];
        b0.y = W[oc * FIN_ + kh + 1];
        b1.x = W[(oc + 16) * FIN_ + kh];
        b1.y = W[(oc + 16) * FIN_ + kh + 1];
        o00 = wmma4(a0, b0, o00);
        o01 = wmma4(a0, b1, o01);
        o10 = wmma4(a1, b0, o10);
        o11 = wmma4(a1, b1, o11);
    }

    // ---- bias + LeakyReLU + store ----
    const float bv0 = bias[oc];
    const float bv1 = bias[oc + 16];
#pragma unroll
    for (int v = 0; v < 8; ++v) {
        const long long row0 = gIdx + mbase + v + 8 * half;
        const long long row1 = row0 + 16;
        float x;
        x = o00[v] + bv0; out[row0 * FOUT_ + oc]      = (x >= 0.f) ? x : 0.01f * x;
        x = o01[v] + bv1; out[row0 * FOUT_ + oc + 16] = (x >= 0.f) ? x : 0.01f * x;
        x = o10[v] + bv0; out[row1 * FOUT_ + oc]      = (x >= 0.f) ? x : 0.01f * x;
        x = o11[v] + bv1; out[row1 * FOUT_ + oc + 16] = (x >= 0.f) ? x : 0.01f * x;
    }
}

// ---------------------------------------------------------------------------
extern "C" void kernel_launch(void* const* d_in, const int* in_sizes, int n_in,
                              void* d_out, int out_size, void* d_ws, size_t ws_size,
                              hipStream_t stream) {
    (void)in_sizes; (void)n_in; (void)out_size; (void)ws_size;
    const float* node = (const float*)d_in[0];   // [B, N, FIN]
    const float* adj  = (const float*)d_in[1];   // [B, N, N]
    const float* W    = (const float*)d_in[2];   // [FOUT, FIN]
    const float* bias = (const float*)d_in[3];   // [FOUT]
    float* out    = (float*)d_out;               // [B, N, FOUT]
    float* invdeg = (float*)d_ws;                // B*N floats (128 KB)

    rowsum_inv_kernel<<<(B_ * N_) / 8, 256, 0, stream>>>(adj, invdeg);
    gcn_fused_kernel<<<(B_ * N_) / ROWS, 256, 0, stream>>>(node, adj, W, bias,
                                                           invdeg, out);
}